// DeformBasicBlock_44195213476552
// MI455X (gfx1250) — compile-verified
//
#include <hip/hip_runtime.h>
#include <math.h>

// Problem dims (fixed by the reference)
#define NN 8
#define CC 64
#define DD 16
#define HH 56
#define WW 56
#define GG 8
#define HWSZ (HH*WW)          // 3136
#define DHW (DD*HWSZ)         // 50176
#define OUT_ELEMS (NN*CC*DHW) // 25690112
#define OFF_ELEMS (NN*GG*DHW) // 3211264
#define RED_N (NN*DHW)        // 401408 elements per channel for BN

typedef __attribute__((ext_vector_type(2))) float v2f;
typedef __attribute__((ext_vector_type(8))) float v8f;

// ---------------------------------------------------------------------------
// Implicit-GEMM core: 16(co) x 16(w) tile, K = 64 ci over 27 taps, via
// V_WMMA_F32_16X16X4_F32 (D = A(16x4) * B(4x16) + C).
//   A fragment: lane m = L%16, k = (L/16)*2 + j.
//   B fragment: lane n = L%16, k = (L/16) + 2*j.
// Weights packed per-lane contiguous: wpack[cot][tap][lane][cc(16)][2]
//   -> 8 x b128 loads per tap cover all 16 K-chunks.
// LDS tile: g[kd(3)][cc(16)][hh(4)][ww(18)][slot(4)] with slot order
//   (k0,k2,k1,k3) -> each lane's B pair is one ds_load_b64.
// ---------------------------------------------------------------------------
__device__ __attribute__((always_inline)) v8f conv_tile_gemm(
    const float* __restrict__ wpack, const float* lds,
    int co_tile, int h_row, int lane, v8f c)
{
  const int slotoff = (lane >> 4) * 2;  // lanes 0-15 -> slots {0,1}, 16-31 -> {2,3}
  const int nW = lane & 15;             // output w within tile
  for (int kd = 0; kd < 3; ++kd)
    for (int kh = 0; kh < 3; ++kh)
      for (int kw = 0; kw < 3; ++kw) {
        const int tap = (kd * 3 + kh) * 3 + kw;
        const float* ap = wpack + ((co_tile * 27 + tap) * 32 + lane) * 32;
        __builtin_prefetch((const void*)(ap + 1024), 0, 0);  // next tap's A block
        float4 a4[8];
#pragma unroll
        for (int q = 0; q < 8; ++q) a4[q] = ((const float4*)ap)[q];
        const float* af = (const float*)a4;
        const int hh = h_row + kh;                     // 0..3
        const int wb = nW + kw;                        // 0..17
        const float* bb = lds + (kd * 64 + hh) * 72 + wb * 4 + slotoff;
#pragma unroll
        for (int cc = 0; cc < 16; ++cc) {
          v2f B = *(const v2f*)(bb + cc * 288);        // ds_load_b64
          v2f A;
          A[0] = af[cc * 2];
          A[1] = af[cc * 2 + 1];
          c = __builtin_amdgcn_wmma_f32_16x16x4_f32(
                  false, A, false, B, (short)0, c, false, false);
        }
      }
  return c;
}

// slot <-> k permutation (self-inverse): 0<->0, 1<->2, 2<->1, 3<->3
__device__ __attribute__((always_inline)) int slot_perm(int s) {
  return (s & 1) * 2 + (s >> 1);
}

// ---------------------------------------------------------------------------
// Repack w[co][ci][kd][kh][kw] (64x64x27) into WMMA A-fragment order:
// wp[((cot*27+tap)*32 + lane)*32 + cc*2 + j], co = cot*16+(lane&15),
// ci = cc*4 + (lane>>4)*2 + j.
// ---------------------------------------------------------------------------
__global__ __launch_bounds__(256) void pack_weights(
    const float* __restrict__ w, float* __restrict__ wp)
{
  int t = blockIdx.x * 256 + threadIdx.x;       // (cot,tap,lane,cc)
  if (t >= 4 * 27 * 32 * 16) return;
  int cc   = t & 15;
  int lane = (t >> 4) & 31;
  int tap  = (t >> 9) % 27;
  int cot  = t / (512 * 27);
  int co   = cot * 16 + (lane & 15);
  int kb   = (lane >> 4) << 1;
#pragma unroll
  for (int j = 0; j < 2; ++j) {
    int ci = cc * 4 + kb + j;
    wp[(((cot * 27 + tap) * 32 + lane) * 16 + cc) * 2 + j] =
        w[(co * CC + ci) * 27 + tap];
  }
}

// ---------------------------------------------------------------------------
// Offset conv: each thread computes all 8 group outputs for one (n,d,h,w).
// Weight index is wave-uniform -> scalar loads; x loaded once per (ci,tap).
// ---------------------------------------------------------------------------
__global__ __launch_bounds__(256) void conv_off_kernel(
    const float* __restrict__ x, const float* __restrict__ w_off,
    const float* __restrict__ b_off, float* __restrict__ off)
{
  int t = blockIdx.x * 256 + threadIdx.x;
  if (t >= NN * DHW) return;
  int n  = t / DHW;
  int sp = t - n * DHW;
  int d  = sp / HWSZ;
  int r  = sp - d * HWSZ;
  int h  = r / WW, w = r - h * WW;
  float acc[GG];
#pragma unroll
  for (int g = 0; g < GG; ++g) acc[g] = b_off[g];
  for (int kd = 0; kd < 3; ++kd) {
    int dd = d + kd - 1; if ((unsigned)dd >= (unsigned)DD) continue;
    for (int kh = 0; kh < 3; ++kh) {
      int hh = h + kh - 1; if ((unsigned)hh >= (unsigned)HH) continue;
      for (int kw = 0; kw < 3; ++kw) {
        int ww = w + kw - 1; if ((unsigned)ww >= (unsigned)WW) continue;
        int tap = (kd * 3 + kh) * 3 + kw;
        const float* xp = x + (n * CC * DD + dd) * HWSZ + hh * WW + ww;
        for (int ci = 0; ci < CC; ++ci) {
          float xv = xp[ci * DHW];
#pragma unroll
          for (int g = 0; g < GG; ++g)
            acc[g] += w_off[(g * CC + ci) * 27 + tap] * xv;
        }
      }
    }
  }
#pragma unroll
  for (int g = 0; g < GG; ++g)
    off[((n * GG + g) * DD + d) * HWSZ + h * WW + w] = acc[g];
}

// ---------------------------------------------------------------------------
// Deformable conv (WMMA): block = (n, d, 2 h-rows, 16 w-cols), 8 waves.
// Phase 1: fill LDS (permuted layout) with D-interpolated gather.
// Phase 2: each wave does a 16co x 16w implicit GEMM (432 WMMA).
// ---------------------------------------------------------------------------
__global__ __launch_bounds__(256) void deform_conv_wmma(
    const float* __restrict__ x, const float* __restrict__ off,
    const float* __restrict__ wpack, float* __restrict__ out1)
{
  __shared__ float g[3 * 16 * 4 * 18 * 4];  // 13824 floats
  int bid = blockIdx.x;
  int wt  = bid & 3;
  int tmp = bid >> 2;
  int ht  = tmp % 28;
  tmp /= 28;
  int d   = tmp & 15;
  int n   = tmp >> 4;
  int h0 = ht * 2, w0 = wt * 16;

  // t decodes the permuted LDS order directly -> contiguous LDS stores.
  for (int t = threadIdx.x; t < 13824; t += 256) {
    int kd = t / 4608;
    int r  = t - kd * 4608;
    int cc = r / 288;
    int r2 = r - cc * 288;
    int hh = r2 / 72;
    int r3 = r2 - hh * 72;
    int ww = r3 >> 2;
    int sl = r3 & 3;
    int ci = cc * 4 + slot_perm(sl);
    int hp = h0 - 1 + hh, wp = w0 - 1 + ww;
    float val = 0.f;
    if ((unsigned)hp < (unsigned)HH && (unsigned)wp < (unsigned)WW) {
      int gi = ci >> 3;
      float ofv = off[((n * GG + gi) * DD + d) * HWSZ + hp * WW + wp];
      float pos = (float)d + ofv + (float)(kd - 1);
      float f0  = floorf(pos);
      float fr  = pos - f0;
      int   i0  = (int)f0;
      const float* xc = x + ((n * CC + ci) * DD) * HWSZ + hp * WW + wp;
      float v0 = ((unsigned)i0       < (unsigned)DD) ? xc[i0 * HWSZ]       : 0.f;
      float v1 = ((unsigned)(i0 + 1) < (unsigned)DD) ? xc[(i0 + 1) * HWSZ] : 0.f;
      val = v0 * (1.f - fr) + v1 * fr;
    }
    g[t] = val;
  }
  __syncthreads();

  int lane    = threadIdx.x & 31;
  int wid     = threadIdx.x >> 5;
  int co_tile = wid & 3;
  int h_row   = wid >> 2;
  v8f c = {};
  c = conv_tile_gemm(wpack, g, co_tile, h_row, lane, c);

  int wv = w0 + (lane & 15);
  int h  = h0 + h_row;
  if (wv < WW) {
    int mbase = co_tile * 16 + ((lane >> 4) << 3);
    float* o = out1 + ((n * CC + mbase) * DD + d) * HWSZ + h * WW + wv;
#pragma unroll
    for (int r = 0; r < 8; ++r) o[r * DHW] = c[r];
  }
}

// ---------------------------------------------------------------------------
// Conv2 (WMMA): same GEMM core; input is BN1+ReLU(out1), fused into LDS fill.
// ---------------------------------------------------------------------------
__global__ __launch_bounds__(256) void conv2_wmma(
    const float* __restrict__ in, const float* __restrict__ scale1,
    const float* __restrict__ bias1, const float* __restrict__ b2,
    const float* __restrict__ wpack, float* __restrict__ out2)
{
  __shared__ float g[3 * 16 * 4 * 18 * 4];
  int bid = blockIdx.x;
  int wt  = bid & 3;
  int tmp = bid >> 2;
  int ht  = tmp % 28;
  tmp /= 28;
  int d   = tmp & 15;
  int n   = tmp >> 4;
  int h0 = ht * 2, w0 = wt * 16;

  for (int t = threadIdx.x; t < 13824; t += 256) {
    int dd = t / 4608;          // kd slice selects input depth d+dd-1
    int r  = t - dd * 4608;
    int cc = r / 288;
    int r2 = r - cc * 288;
    int hh = r2 / 72;
    int r3 = r2 - hh * 72;
    int ww = r3 >> 2;
    int sl = r3 & 3;
    int ci = cc * 4 + slot_perm(sl);
    int hp = h0 - 1 + hh, wp = w0 - 1 + ww;
    int dp = d + dd - 1;
    float val = 0.f;
    if ((unsigned)hp < (unsigned)HH && (unsigned)wp < (unsigned)WW &&
        (unsigned)dp < (unsigned)DD) {
      float v = in[((n * CC + ci) * DD + dp) * HWSZ + hp * WW + wp];
      val = fmaxf(scale1[ci] * v + bias1[ci], 0.f);
    }
    g[t] = val;
  }
  __syncthreads();

  int lane    = threadIdx.x & 31;
  int wid     = threadIdx.x >> 5;
  int co_tile = wid & 3;
  int h_row   = wid >> 2;
  int mbase   = co_tile * 16 + ((lane >> 4) << 3);
  v8f c;
#pragma unroll
  for (int r = 0; r < 8; ++r) c[r] = b2[mbase + r];
  c = conv_tile_gemm(wpack, g, co_tile, h_row, lane, c);

  int wv = w0 + (lane & 15);
  int h  = h0 + h_row;
  if (wv < WW) {
    float* o = out2 + ((n * CC + mbase) * DD + d) * HWSZ + h * WW + wv;
#pragma unroll
    for (int r = 0; r < 8; ++r) o[r * DHW] = c[r];
  }
}

// ---------------------------------------------------------------------------
// BN stats, deterministic two-stage: stage 1 block (c, n) tree-reduces 50176
// elements; stage 2 folds 8 partials and gamma/beta into scale/bias.
// ---------------------------------------------------------------------------
__global__ __launch_bounds__(256) void bn_stats_partial(
    const float* __restrict__ p, float* __restrict__ psum, float* __restrict__ psq)
{
  int c = blockIdx.x, n = blockIdx.y;
  const float* base = p + (n * CC + c) * DHW;
  float s = 0.f, s2 = 0.f;
  for (int i = threadIdx.x; i < DHW; i += 256) {
    float v = base[i];
    s += v; s2 += v * v;
  }
  __shared__ float ls[256], ls2[256];
  ls[threadIdx.x] = s; ls2[threadIdx.x] = s2;
  __syncthreads();
  for (int st = 128; st > 0; st >>= 1) {
    if (threadIdx.x < st) {
      ls[threadIdx.x]  += ls[threadIdx.x + st];
      ls2[threadIdx.x] += ls2[threadIdx.x + st];
    }
    __syncthreads();
  }
  if (threadIdx.x == 0) { psum[c * NN + n] = ls[0]; psq[c * NN + n] = ls2[0]; }
}

__global__ void bn_finalize(
    const float* __restrict__ psum, const float* __restrict__ psq,
    const float* __restrict__ gamma, const float* __restrict__ beta,
    float* __restrict__ scale, float* __restrict__ bias)
{
  int c = threadIdx.x;
  if (c >= CC) return;
  float S = 0.f, S2 = 0.f;
  for (int n = 0; n < NN; ++n) { S += psum[c * NN + n]; S2 += psq[c * NN + n]; }
  const float inv = 1.f / (float)RED_N;
  float mean = S * inv;
  float var  = S2 * inv - mean * mean;
  float sc   = gamma[c] * rsqrtf(var + 1e-5f);
  scale[c] = sc;
  bias[c]  = beta[c] - mean * sc;
}

// ---------------------------------------------------------------------------
// Final: out = relu(bn2(out2) + x), in place on d_out seg0, float4 traffic.
// ---------------------------------------------------------------------------
__global__ __launch_bounds__(256) void residual_final(
    float* __restrict__ out, const float* __restrict__ x,
    const float* __restrict__ scale, const float* __restrict__ bias)
{
  int i4 = blockIdx.x * 256 + threadIdx.x;
  if (i4 >= OUT_ELEMS / 4) return;
  int c = (i4 / (DHW / 4)) & (CC - 1);
  float4 o  = ((const float4*)out)[i4];
  float4 xx = ((const float4*)x)[i4];
  float sc = scale[c], bi = bias[c];
  o.x = fmaxf(sc * o.x + bi + xx.x, 0.f);
  o.y = fmaxf(sc * o.y + bi + xx.y, 0.f);
  o.z = fmaxf(sc * o.z + bi + xx.z, 0.f);
  o.w = fmaxf(sc * o.w + bi + xx.w, 0.f);
  ((float4*)out)[i4] = o;
}

extern "C" void kernel_launch(void* const* d_in, const int* in_sizes, int n_in,
                              void* d_out, int out_size, void* d_ws, size_t ws_size,
                              hipStream_t stream) {
  const float* x     = (const float*)d_in[0];
  const float* w_off = (const float*)d_in[1];
  const float* b_off = (const float*)d_in[2];
  const float* w1    = (const float*)d_in[3];
  const float* w2    = (const float*)d_in[4];
  const float* b2    = (const float*)d_in[5];
  const float* g1    = (const float*)d_in[6];
  const float* be1   = (const float*)d_in[7];
  const float* g2    = (const float*)d_in[8];
  const float* be2   = (const float*)d_in[9];

  float* out = (float*)d_out;          // seg0: main output (also conv2 scratch)
  float* off = out + OUT_ELEMS;        // seg1: offsets output

  float* ws     = (float*)d_ws;
  float* out1   = ws;                              // 25,690,112 f
  float* wpack1 = out1 + OUT_ELEMS;                // 110,592 f
  float* wpack2 = wpack1 + 4 * 27 * 16 * 64;       // 110,592 f
  float* psum   = wpack2 + 4 * 27 * 16 * 64;       // 512 f
  float* psq    = psum + CC * NN;                  // 512 f
  float* scale1 = psq + CC * NN;
  float* bias1  = scale1 + CC;
  float* scale2 = bias1 + CC;
  float* bias2  = scale2 + CC;

  pack_weights<<<216, 256, 0, stream>>>(w1, wpack1);
  pack_weights<<<216, 256, 0, stream>>>(w2, wpack2);

  conv_off_kernel<<<(NN * DHW + 255) / 256, 256, 0, stream>>>(x, w_off, b_off, off);

  deform_conv_wmma<<<NN * DD * 28 * 4, 256, 0, stream>>>(x, off, wpack1, out1);

  bn_stats_partial<<<dim3(CC, NN), 256, 0, stream>>>(out1, psum, psq);
  bn_finalize<<<1, 64, 0, stream>>>(psum, psq, g1, be1, scale1, bias1);

  conv2_wmma<<<NN * DD * 28 * 4, 256, 0, stream>>>(out1, scale1, bias1, b2, wpack2, out);

  bn_stats_partial<<<dim3(CC, NN), 256, 0, stream>>>(out, psum, psq);
  bn_finalize<<<1, 64, 0, stream>>>(psum, psq, g2, be2, scale2, bias2);

  residual_final<<<(OUT_ELEMS / 4 + 255) / 256, 256, 0, stream>>>(out, x, scale2, bias2);
}